// GAT_17970143167222
// MI455X (gfx1250) — compile-verified
//
#include <hip/hip_runtime.h>
#include <hip/hip_bf16.h>
#include <math.h>

// ---------------------------------------------------------------------------
// GAT (2x GATConv) on MI455X / gfx1250.
// Layer 1 (in=1, H=8, C=8, concat) collapses algebraically:
//   h1[n,k]      = x[n] * w1[k]
//   alpha[e,h]   = lrelu( x[s]*S1[h] + x[d]*D1[h] + ea[e]*E1[h] )
//   out1[n,h,c]  = w1[h*8+c] * T[n,h],   T[n,h] = sum_e x[s]*attn[e,h]
// Layer 2 (in=64, H=1, C=3): dense h1_elu @ W2 done with V_WMMA_F32_16X16X4_F32.
// Segment softmax = 3 edge passes/layer (atomic max, exp-sum, weighted scatter),
// logits recomputed each pass from register-resident coefficients.
// ELU is branchless (max(a,0) + expm1(min(a,0))) so EXEC stays all-ones
// through the entire WMMA accumulation region (ISA 7.12 requirement).
// ---------------------------------------------------------------------------

#define LRELU_SLOPE 0.2f
#define SOFT_EPS    1e-16f

typedef __attribute__((ext_vector_type(2))) float v2f;
typedef __attribute__((ext_vector_type(8))) float v8f;

// order-preserving float <-> uint mapping for atomicMax on floats
__device__ __forceinline__ unsigned mapf(float f) {
  unsigned u = __float_as_uint(f);
  return (u & 0x80000000u) ? ~u : (u | 0x80000000u);
}
__device__ __forceinline__ float unmapf(unsigned u) {
  return (u & 0x80000000u) ? __uint_as_float(u ^ 0x80000000u)
                           : __uint_as_float(~u);
}

// branchless ELU: exact (exp(0)-1 == 0 for the non-negative side)
__device__ __forceinline__ float elu_nb(float a) {
  return fmaxf(a, 0.f) + (expf(fminf(a, 0.f)) - 1.f);
}

// ---------------------------------------------------------------------------
// K0: fold weight/attention vectors into per-head coefficients.
// coef layout (floats): [0..7]=S1, [8..15]=D1, [16..23]=E1, [24]=E2
// ---------------------------------------------------------------------------
__global__ void k0_coefs(const float* __restrict__ w1,
                         const float* __restrict__ we1,
                         const float* __restrict__ as1,
                         const float* __restrict__ ad1,
                         const float* __restrict__ ae1,
                         const float* __restrict__ we2,
                         const float* __restrict__ ae2,
                         float* __restrict__ coef) {
  int h = threadIdx.x;
  if (h < 8) {
    float s = 0.f, d = 0.f, e = 0.f;
    #pragma unroll
    for (int c = 0; c < 8; ++c) {
      float wk = w1[h * 8 + c];
      s += wk * as1[h * 8 + c];
      d += wk * ad1[h * 8 + c];
      e += we1[h * 8 + c] * ae1[h * 8 + c];
    }
    coef[h] = s; coef[8 + h] = d; coef[16 + h] = e;
  }
  if (threadIdx.x == 8) {
    float e2 = 0.f;
    #pragma unroll
    for (int c = 0; c < 3; ++c) e2 += we2[c] * ae2[c];
    coef[24] = e2;
  }
}

// ---------------------------------------------------------------------------
// K1: initialize accumulators (ws is poisoned 0xAA; d_out too).
// ---------------------------------------------------------------------------
__global__ void k1_init(float* __restrict__ T, float* __restrict__ den1,
                        unsigned* __restrict__ amax1,
                        float* __restrict__ den2, unsigned* __restrict__ amax2,
                        float* __restrict__ out, int n_nodes) {
  int i = blockIdx.x * blockDim.x + threadIdx.x;   // covers n_nodes*8
  if (i < n_nodes * 8) { T[i] = 0.f; den1[i] = 0.f; amax1[i] = 0u; }
  if (i < n_nodes)     { den2[i] = 0.f; amax2[i] = 0u; }
  if (i < n_nodes * 3) { out[i] = 0.f; }
}

// ---------------------------------------------------------------------------
// Layer-1 edge passes (H=8). phase 0: amax, 1: denom, 2: weighted scatter to T
// ---------------------------------------------------------------------------
template <int PHASE>
__global__ void k2_edges_l1(const int* __restrict__ ei, int n_edges,
                            const float* __restrict__ x,
                            const float* __restrict__ ea,
                            const float* __restrict__ coef,
                            unsigned* __restrict__ amax1,
                            float* __restrict__ den1,
                            float* __restrict__ T) {
  int e = blockIdx.x * blockDim.x + threadIdx.x;
  if (e >= n_edges) return;
  int s = ei[e];
  int d = ei[n_edges + e];
  float xs = x[s], xd = x[d], w = ea[e];
  #pragma unroll
  for (int h = 0; h < 8; ++h) {
    float a = xs * coef[h] + xd * coef[8 + h] + w * coef[16 + h];
    a = (a > 0.f) ? a : LRELU_SLOPE * a;
    int idx = d * 8 + h;
    if (PHASE == 0) {
      atomicMax(&amax1[idx], mapf(a));
    } else {
      float ex = expf(a - unmapf(amax1[idx]));
      if (PHASE == 1) {
        atomicAdd(&den1[idx], ex);
      } else {
        float attn = ex / (den1[idx] + SOFT_EPS);
        atomicAdd(&T[idx], xs * attn);
      }
    }
  }
}

// ---------------------------------------------------------------------------
// K3: per 16-node tile (one wave32): h1_elu(16x64) @ W2pad(64x16) via
// 16x V_WMMA_F32_16X16X4_F32 (K=64). Only columns 0..2 of D are real.
// A layout (ISA 7.12.2, 32-bit A 16x4): lanes 0-15 -> M=lane, K={0,1};
// lanes 16-31 -> M=lane-16, K={2,3}. B mirrors it transposed. C/D: VGPR v,
// lanes 0-15 -> M=v, N=lane; lanes 16-31 -> M=v+8, N=lane-16.
// All guards are wave-uniform or load-side clamps: EXEC is all-ones at every
// WMMA issue point.
// ---------------------------------------------------------------------------
__global__ void k3_linear_wmma(const float* __restrict__ T,
                               const float* __restrict__ w1,
                               const float* __restrict__ b1,
                               const float* __restrict__ w2,
                               float* __restrict__ h2, int n_nodes) {
  int lane = threadIdx.x & 31;
  int wave = threadIdx.x >> 5;
  int tile = blockIdx.x * (blockDim.x >> 5) + wave;
  int base = tile * 16;
  if (base >= n_nodes) return;            // wave-uniform: EXEC stays all-ones
  int half = lane >> 4;                   // 0|1 -> K pair select
  int mloc = lane & 15;                   // row within tile / output column
  int mclamp = min(base + mloc, n_nodes - 1);
  bool bcol = (mloc < 3);

  v8f acc = {};
  #pragma unroll
  for (int kk = 0; kk < 16; ++kk) {
    int k0 = kk * 4 + half * 2;
    // A = elu(w1[k]*T[m, k>>3] + b1[k])  (branchless; no EXEC changes)
    float t0 = T[(size_t)mclamp * 8 + (k0 >> 3)];
    float t1 = T[(size_t)mclamp * 8 + ((k0 + 1) >> 3)];
    float a0 = elu_nb(w1[k0] * t0 + b1[k0]);
    float a1 = elu_nb(w1[k0 + 1] * t1 + b1[k0 + 1]);
    v2f A; A.x = a0; A.y = a1;
    // B = W2 zero-padded from (64,3) to (64,16); select is branchless cndmask
    v2f B;
    B.x = bcol ? w2[k0 * 3 + mloc] : 0.f;
    B.y = bcol ? w2[(k0 + 1) * 3 + mloc] : 0.f;
    acc = __builtin_amdgcn_wmma_f32_16x16x4_f32(
        false, A, false, B, (short)0, acc, false, false);
  }
  #pragma unroll
  for (int v = 0; v < 8; ++v) {
    int node = base + v + 8 * half;
    if (node < n_nodes && bcol) h2[(size_t)node * 3 + mloc] = acc[v];
  }
}

// ---------------------------------------------------------------------------
// K3b: per-node attention logit halves for layer 2.
// ---------------------------------------------------------------------------
__global__ void k3b_node_alpha(const float* __restrict__ h2,
                               const float* __restrict__ as2,
                               const float* __restrict__ ad2,
                               float* __restrict__ asrc,
                               float* __restrict__ adst, int n_nodes) {
  int n = blockIdx.x * blockDim.x + threadIdx.x;
  if (n >= n_nodes) return;
  float h0 = h2[n * 3 + 0], h1 = h2[n * 3 + 1], hc = h2[n * 3 + 2];
  asrc[n] = h0 * as2[0] + h1 * as2[1] + hc * as2[2];
  adst[n] = h0 * ad2[0] + h1 * ad2[1] + hc * ad2[2];
}

// ---------------------------------------------------------------------------
// Layer-2 edge passes (H=1). phase 0: amax, 1: denom, 2: scatter msg to out.
// ---------------------------------------------------------------------------
template <int PHASE>
__global__ void k4_edges_l2(const int* __restrict__ ei, int n_edges,
                            const float* __restrict__ ea,
                            const float* __restrict__ coef,   // coef[24] = E2
                            const float* __restrict__ asrc,
                            const float* __restrict__ adst,
                            const float* __restrict__ h2,
                            unsigned* __restrict__ amax2,
                            float* __restrict__ den2,
                            float* __restrict__ out) {
  int e = blockIdx.x * blockDim.x + threadIdx.x;
  if (e >= n_edges) return;
  int s = ei[e];
  int d = ei[n_edges + e];
  float a = asrc[s] + adst[d] + ea[e] * coef[24];
  a = (a > 0.f) ? a : LRELU_SLOPE * a;
  if (PHASE == 0) {
    atomicMax(&amax2[d], mapf(a));
  } else {
    float ex = expf(a - unmapf(amax2[d]));
    if (PHASE == 1) {
      atomicAdd(&den2[d], ex);
    } else {
      float attn = ex / (den2[d] + SOFT_EPS);
      atomicAdd(&out[d * 3 + 0], h2[(size_t)s * 3 + 0] * attn);
      atomicAdd(&out[d * 3 + 1], h2[(size_t)s * 3 + 1] * attn);
      atomicAdd(&out[d * 3 + 2], h2[(size_t)s * 3 + 2] * attn);
    }
  }
}

// K5: add output bias in place.
__global__ void k5_bias(float* __restrict__ out, const float* __restrict__ b2,
                        int n_nodes) {
  int i = blockIdx.x * blockDim.x + threadIdx.x;
  if (i < n_nodes * 3) out[i] += b2[i % 3];
}

// ---------------------------------------------------------------------------
extern "C" void kernel_launch(void* const* d_in, const int* in_sizes, int n_in,
                              void* d_out, int out_size, void* d_ws, size_t ws_size,
                              hipStream_t stream) {
  const float* x    = (const float*)d_in[0];
  const int*   ei   = (const int*)d_in[1];
  const float* ea   = (const float*)d_in[2];
  const float* w1   = (const float*)d_in[3];
  const float* we1  = (const float*)d_in[4];
  const float* as1  = (const float*)d_in[5];
  const float* ad1  = (const float*)d_in[6];
  const float* ae1  = (const float*)d_in[7];
  const float* b1   = (const float*)d_in[8];
  const float* w2   = (const float*)d_in[9];
  const float* we2  = (const float*)d_in[10];
  const float* as2  = (const float*)d_in[11];
  const float* ad2  = (const float*)d_in[12];
  const float* ae2  = (const float*)d_in[13];
  const float* b2   = (const float*)d_in[14];
  float* out = (float*)d_out;

  const int n_nodes = in_sizes[0];
  const int n_edges = in_sizes[1] / 2;

  // workspace layout (floats)
  float* ws = (float*)d_ws;
  float*    coef  = ws;                      // 32
  float*    T     = coef + 32;               // 8N
  unsigned* amax1 = (unsigned*)(T + (size_t)8 * n_nodes);          // 8N
  float*    den1  = (float*)(amax1 + (size_t)8 * n_nodes);         // 8N
  float*    h2    = den1 + (size_t)8 * n_nodes;                    // 3N
  float*    asrc  = h2 + (size_t)3 * n_nodes;                      // N
  float*    adst  = asrc + n_nodes;                                // N
  unsigned* amax2 = (unsigned*)(adst + n_nodes);                   // N
  float*    den2  = (float*)(amax2 + n_nodes);                     // N

  const int TB = 256;
  const int gE  = (n_edges + TB - 1) / TB;
  const int gN8 = (n_nodes * 8 + TB - 1) / TB;
  const int gN  = (n_nodes + TB - 1) / TB;
  const int gN3 = (n_nodes * 3 + TB - 1) / TB;
  const int gT  = ((n_nodes + 15) / 16 + (TB / 32) - 1) / (TB / 32); // 16-node tiles, 8 waves/block

  k0_coefs<<<1, 64, 0, stream>>>(w1, we1, as1, ad1, ae1, we2, ae2, coef);
  k1_init<<<gN8, TB, 0, stream>>>(T, den1, amax1, den2, amax2, out, n_nodes);

  k2_edges_l1<0><<<gE, TB, 0, stream>>>(ei, n_edges, x, ea, coef, amax1, den1, T);
  k2_edges_l1<1><<<gE, TB, 0, stream>>>(ei, n_edges, x, ea, coef, amax1, den1, T);
  k2_edges_l1<2><<<gE, TB, 0, stream>>>(ei, n_edges, x, ea, coef, amax1, den1, T);

  k3_linear_wmma<<<gT, TB, 0, stream>>>(T, w1, b1, w2, h2, n_nodes);
  k3b_node_alpha<<<gN, TB, 0, stream>>>(h2, as2, ad2, asrc, adst, n_nodes);

  k4_edges_l2<0><<<gE, TB, 0, stream>>>(ei, n_edges, ea, coef, asrc, adst, h2, amax2, den2, out);
  k4_edges_l2<1><<<gE, TB, 0, stream>>>(ei, n_edges, ea, coef, asrc, adst, h2, amax2, den2, out);
  k4_edges_l2<2><<<gE, TB, 0, stream>>>(ei, n_edges, ea, coef, asrc, adst, h2, amax2, den2, out);

  k5_bias<<<gN3, TB, 0, stream>>>(out, b2, n_nodes);
}